// Linear_86371792322560
// MI455X (gfx1250) — compile-verified
//
#include <hip/hip_runtime.h>
#include <math.h>

// ---------------------------------------------------------------------------
// Social-force pedestrian sim, N=4096.
// Hot loop: O(N^2) pairwise repulsion. Strategy for MI455X (gfx1250):
//   * positions staged once in LDS (32KB << 320KB/WGP)
//   * 16x16 dist^2 tiles via V_WMMA_F32_16X16X4_F32:
//       d2 = |pi|^2 + |pj|^2 - 2 pi.pj + EPS  (rank-4 product, C preloaded EPS)
//     used as a tile/element cull (d2 > 400 -> mag < 1e-16, skip transcendentals)
//   * wave-vote early-out: if no lane of the wave sees d2 < 400 in the tile,
//     the whole 256-element tile is skipped (WMMA acts as the screening pass)
//   * exact dx/dy/d2 recomputed on VALU for surviving pairs (accuracy; the
//     Gram form cancels catastrophically for near pairs at |p| ~ 40)
//   * readfirstlane on the wave id so tile indices / loop control are SALU
//     (keeps EXEC trivially all-ones around the WMMA, as the ISA requires)
//   * wave32: 1024 waves (256 row tiles x 4 column splits), shfl_xor butterfly
//     reduction within 16-lane groups, deterministic LDS combine (no atomics)
// ---------------------------------------------------------------------------

#define N_AG 4096
#define TILE 16
#define NT (N_AG / TILE)        // 256 tiles per dimension
#define COL_PARTS 4
#define TPB 2                    // row tiles per block (8 waves = 2 tiles x 4 parts)
#define EPSF 1e-3f
#define CULL_D2 400.0f           // dist > 20 -> mag = 5*exp(-38.8) ~ 7e-17

typedef float v2f __attribute__((ext_vector_type(2)));
typedef float v8f __attribute__((ext_vector_type(8)));

__global__ __launch_bounds__(256) void sfm_forces_kernel(
    const float* __restrict__ state,   // [N,4] px,py,vx,vy
    const float* __restrict__ goals,   // [N,2]
    float* __restrict__ out_state)     // [N,4] px',py',vx',vy'
{
    __shared__ float s_px[N_AG];                         // 16 KB
    __shared__ float s_py[N_AG];                         // 16 KB
    __shared__ float s_part[TPB * COL_PARTS * TILE * 2]; // 1 KB partial rf

    const int t = threadIdx.x;

    // Cooperative load of all positions into LDS (state row = float4, pos = .xy)
    for (int k = t; k < N_AG; k += 256) {
        float2 p = ((const float2*)state)[2 * k];
        s_px[k] = p.x;
        s_py[k] = p.y;
    }
    __syncthreads();

    // Wave id is wave-uniform: force it scalar so tile indices and the j-loop
    // are SALU-controlled (no exec-mask loop, EXEC all-ones at the WMMA).
    const int wid  = __builtin_amdgcn_readfirstlane(t >> 5);   // 0..7
    const int lane = t & 31;
    const int lh   = lane >> 4;    // lane half (0 or 1)
    const int n    = lane & 15;    // column within tile (C/D layout: N = lane%16)

    const int tib = wid >> 2;                       // row tile in block: 0..1
    const int cp  = wid & 3;                        // column partition: 0..3
    const int i0  = (blockIdx.x * TPB + tib) * TILE; // scalar

    // Row data for the 8 M-rows this lane owns in the C/D layout: M = v + 8*lh
    float pix[8], piy[8];
    #pragma unroll
    for (int v = 0; v < 8; ++v) {
        pix[v] = s_px[i0 + v + 8 * lh];
        piy[v] = s_py[i0 + v + 8 * lh];
    }

    // A fragment (16x4, row M = lane%16, K = vgpr + 2*lh):
    //   A[M] = [pix, piy, |pi|^2, 1]
    const float apx = s_px[i0 + n];
    const float apy = s_py[i0 + n];
    const float ar  = apx * apx + apy * apy;
    v2f a;
    a.x = lh ? ar   : apx;   // K=0 (half0) / K=2 (half1)
    a.y = lh ? 1.0f : apy;   // K=1 (half0) / K=3 (half1)

    float ax[8] = {0.f, 0.f, 0.f, 0.f, 0.f, 0.f, 0.f, 0.f};
    float ay[8] = {0.f, 0.f, 0.f, 0.f, 0.f, 0.f, 0.f, 0.f};

    for (int jt = cp; jt < NT; jt += COL_PARTS) {   // scalar loop
        const int j0 = jt * TILE;                   // scalar

        // This lane's column j data (same for both halves)
        const float pjx = s_px[j0 + n];
        const float pjy = s_py[j0 + n];
        const float rj  = pjx * pjx + pjy * pjy;

        // B fragment (4x16, col N = lane%16, K = vgpr + 2*lh):
        //   B col j = [-2 pjx, -2 pjy, 1, |pj|^2]
        v2f b;
        b.x = lh ? 1.0f : -2.0f * pjx;  // K=0 / K=2
        b.y = lh ? rj   : -2.0f * pjy;  // K=1 / K=3

        v8f c = {EPSF, EPSF, EPSF, EPSF, EPSF, EPSF, EPSF, EPSF};
        // d2w[v] ~= |pi|^2 + |pj|^2 - 2 pi.pj + EPS   for M=v+8*lh, N=n
        v8f d2w = __builtin_amdgcn_wmma_f32_16x16x4_f32(
            false, a, false, b, (short)0, c, false, false);

        // Tile-level screen: if no element anywhere in the wave is within the
        // interaction radius, skip the whole tile (w==0 contributes nothing).
        // Diagonal tiles always pass (d2 == EPS on the diagonal).
        bool near_any = false;
        #pragma unroll
        for (int v = 0; v < 8; ++v) near_any |= (d2w[v] < CULL_D2);
        if (__any(near_any)) {
            const bool diag = (j0 == i0);   // scalar
            #pragma unroll
            for (int v = 0; v < 8; ++v) {
                const float dx = pix[v] - pjx;
                const float dy = piy[v] - pjy;
                float w = 0.0f;
                if (d2w[v] < CULL_D2) {
                    // exact d2 on VALU for accuracy on near pairs
                    const float d2   = fmaf(dx, dx, fmaf(dy, dy, EPSF));
                    const float dist = sqrtf(d2);
                    w = 5.0f * __expf((0.6f - dist) * 2.0f) / dist;  // mag/dist
                    if (diag && (v + 8 * lh) == n) w = 0.0f;          // i == j
                }
                ax[v] = fmaf(w, dx, ax[v]);
                ay[v] = fmaf(w, dy, ay[v]);
            }
        }
    }

    // Butterfly reduction over the 16 lanes sharing each row group
    // (xor with 1,2,4,8 stays inside each 16-lane half)
    #pragma unroll
    for (int off = 1; off < 16; off <<= 1) {
        #pragma unroll
        for (int v = 0; v < 8; ++v) {
            ax[v] += __shfl_xor(ax[v], off, 32);
            ay[v] += __shfl_xor(ay[v], off, 32);
        }
    }

    if (n == 0) {   // lanes 0 and 16 hold the group sums for rows 0-7 / 8-15
        #pragma unroll
        for (int v = 0; v < 8; ++v) {
            const int row = v + 8 * lh;
            s_part[((tib * COL_PARTS + cp) * TILE + row) * 2 + 0] = ax[v];
            s_part[((tib * COL_PARTS + cp) * TILE + row) * 2 + 1] = ay[v];
        }
    }
    __syncthreads();

    // Finalize 32 agents per block: fixed-order combine (deterministic), then
    // the O(N) state update.
    if (t < TPB * TILE) {
        const int tib2 = t / TILE;
        const int row  = t % TILE;
        const int i    = blockIdx.x * (TPB * TILE) + t;

        float rfx = 0.0f, rfy = 0.0f;
        #pragma unroll
        for (int p = 0; p < COL_PARTS; ++p) {
            rfx += s_part[((tib2 * COL_PARTS + p) * TILE + row) * 2 + 0];
            rfy += s_part[((tib2 * COL_PARTS + p) * TILE + row) * 2 + 1];
        }

        const float px = s_px[i], py = s_py[i];
        const float vx = state[4 * i + 2], vy = state[4 * i + 3];
        const float gx = goals[2 * i], gy = goals[2 * i + 1];

        const float tgx = gx - px, tgy = gy - py;
        const float dg  = sqrtf(tgx * tgx + tgy * tgy + EPSF);
        const float smax = (i == 0) ? 1.5f : 1.0f;   // robot is agent 0
        const float vdx = tgx / dg * smax;
        const float vdy = tgy / dg * smax;

        const float Fx = rfx + 120.0f * (vdx - vx);  // K*PED_MASS = 2*60
        const float Fy = rfy + 120.0f * (vdy - vy);

        float vnx = vx + Fx * (0.2f / 60.0f);        // DT / PED_MASS
        float vny = vy + Fy * (0.2f / 60.0f);
        const float sp = sqrtf(vnx * vnx + vny * vny + EPSF);
        const float sc = fminf(1.0f, smax / sp);
        vnx *= sc; vny *= sc;

        float4 o;
        o.x = px + vnx * 0.2f;
        o.y = py + vny * 0.2f;
        o.z = vnx;
        o.w = vny;
        ((float4*)out_state)[i] = o;
    }
}

// Cost kernel: needs pos_new[0] (robot), so runs after the forces kernel.
__global__ __launch_bounds__(256) void sfm_cost_kernel(
    const float* __restrict__ out_state,
    const float* __restrict__ cost_in,
    const float* __restrict__ goals,
    const float* __restrict__ robot_init,
    float* __restrict__ cost_out)
{
    const int i = blockIdx.x * 256 + threadIdx.x;
    if (i >= N_AG) return;

    const float rx = out_state[0], ry = out_state[1];       // robot pos_new
    const float gx = goals[0],     gy = goals[1];           // robot goal
    const float ix = robot_init[0], iy = robot_init[1];

    const float d1x = gx - ix, d1y = gy - iy;
    const float d2x = gx - rx, d2y = gy - ry;
    const float pg = sqrtf(d1x * d1x + d1y * d1y + EPSF)
                   - sqrtf(d2x * d2x + d2y * d2y + EPSF);

    const float dx = out_state[4 * i + 0] - rx;
    const float dy = out_state[4 * i + 1] - ry;
    const float dr2 = dx * dx + dy * dy;

    // A_COST * exp(-dr2 / (2*B_COST^2)) - pg ; 2*B^2 = 8
    const float temp = 3.0f * __expf(-dr2 * 0.125f) - pg;
    cost_out[i] = cost_in[i] + temp;
}

extern "C" void kernel_launch(void* const* d_in, const int* in_sizes, int n_in,
                              void* d_out, int out_size, void* d_ws, size_t ws_size,
                              hipStream_t stream) {
    // setup_inputs order: state [N,4], cost [N,1], goals [N,2], robot_init_pose [2]
    const float* state = (const float*)d_in[0];
    const float* cost  = (const float*)d_in[1];
    const float* goals = (const float*)d_in[2];
    const float* rinit = (const float*)d_in[3];

    float* out_state = (float*)d_out;            // first N*4 floats
    float* out_cost  = (float*)d_out + N_AG * 4; // next  N   floats

    // 256 row tiles / 2 per block = 128 blocks x 8 waves = 1024 waves
    sfm_forces_kernel<<<NT / TPB, 256, 0, stream>>>(state, goals, out_state);
    sfm_cost_kernel<<<(N_AG + 255) / 256, 256, 0, stream>>>(
        out_state, cost, goals, rinit, out_cost);
}